// _MCELoss_55662776156557
// MI455X (gfx1250) — compile-verified
//
#include <hip/hip_runtime.h>
#include <hip/hip_bf16.h>

#define N_BINS 15
#define C 128
#define WAVES_PER_BLOCK 8
#define THREADS (WAVES_PER_BLOCK * 32)
#define MAIN_BLOCKS 4096

// ---- feature detection ------------------------------------------------------
#if defined(__has_builtin)
#  if __has_builtin(__builtin_amdgcn_global_load_async_to_lds_b128)
#    define HAVE_ASYNC_B128 1
#  endif
#  if __has_builtin(__builtin_amdgcn_s_wait_asynccnt)
#    define HAVE_WAIT_ASYNC 1
#  endif
#  if __has_builtin(__builtin_amdgcn_exp2f)
#    define EXP2F __builtin_amdgcn_exp2f
#  endif
#endif
#ifndef EXP2F
#  define EXP2F exp2f
#endif

#if defined(HAVE_ASYNC_B128)
#  if defined(HAVE_WAIT_ASYNC)
#    define WAIT_ASYNC(N_) __builtin_amdgcn_s_wait_asynccnt(N_)
#  else
#    define WAIT_ASYNC(N_) asm volatile("s_wait_asynccnt " #N_ ::: "memory")
#  endif
// builtin signature (from hipcc diagnostic): params are v4i* in AS1 / AS3
typedef int v4i __attribute__((ext_vector_type(4)));
typedef __attribute__((address_space(1))) v4i* gas_v4i;
typedef __attribute__((address_space(3))) v4i* las_v4i;
#endif

// ws layout: float[3*N_BINS] = {count, sum_conf, sum_acc} interleaved per bin.

__global__ void mce_init_ws(float* __restrict__ ws) {
    int i = threadIdx.x;
    if (i < 3 * N_BINS) ws[i] = 0.0f;
}

__global__ __launch_bounds__(THREADS) void mce_main(
        const float* __restrict__ logits, const int* __restrict__ labels,
        float* __restrict__ ws, int n)
{
    __shared__ float hist[3 * N_BINS];
#if defined(HAVE_ASYNC_B128)
    __shared__ __align__(16) float lbuf[WAVES_PER_BLOCK][2][C];   // 8 KB double buffers
#endif

    const int lane = threadIdx.x & 31;
    const int wave = threadIdx.x >> 5;
    const int gw   = blockIdx.x * WAVES_PER_BLOCK + wave;   // global wave id
    const int nw   = gridDim.x * WAVES_PER_BLOCK;           // total waves

    for (int i = threadIdx.x; i < 3 * N_BINS; i += blockDim.x) hist[i] = 0.0f;
    __syncthreads();

    const int base = lane << 2;          // first column this lane owns
    const float K = 1.4426950408889634f; // log2(e)

#if defined(HAVE_ASYNC_B128)
    int p = 0;
    if (gw < n) {
        __builtin_amdgcn_global_load_async_to_lds_b128(
            (gas_v4i)(logits + (size_t)gw * C + base),
            (las_v4i)(&lbuf[wave][0][base]), 0, 0);
    }
    for (int row = gw; row < n; row += nw) {
        const int next = row + nw;
        if (next < n) {
            __builtin_amdgcn_global_load_async_to_lds_b128(
                (gas_v4i)(logits + (size_t)next * C + base),
                (las_v4i)(&lbuf[wave][p ^ 1][base]), 0, 0);
            WAIT_ASYNC(1);               // current buffer's copy is the older one
        } else {
            WAIT_ASYNC(0);
        }
        asm volatile("" ::: "memory");   // keep ds reads after the wait
        float4 x = ((const float4*)lbuf[wave][p])[lane];
        p ^= 1;
#else
    for (int row = gw; row < n; row += nw) {
        float4 x = ((const float4*)(logits + (size_t)row * C))[lane];
#endif
        // per-lane max + argmax (first occurrence wins: strict >)
        float m = x.x; int idx = base;
        if (x.y > m) { m = x.y; idx = base + 1; }
        if (x.z > m) { m = x.z; idx = base + 2; }
        if (x.w > m) { m = x.w; idx = base + 3; }
        // wave32 butterfly: max, tie-break to lowest index
        #pragma unroll
        for (int off = 16; off; off >>= 1) {
            float om = __shfl_xor(m, off, 32);
            int   oi = __shfl_xor(idx, off, 32);
            if (om > m || (om == m && oi < idx)) { m = om; idx = oi; }
        }
        // conf = 1 / sum_j exp(x_j - max)
        float s = EXP2F(K * (x.x - m)) + EXP2F(K * (x.y - m)) +
                  EXP2F(K * (x.z - m)) + EXP2F(K * (x.w - m));
        #pragma unroll
        for (int off = 16; off; off >>= 1) s += __shfl_xor(s, off, 32);
        const float conf = 1.0f / s;
        const float acc  = (idx == labels[row]) ? 1.0f : 0.0f;

        // searchsorted(uppers, conf, 'left'), uppers[j] = (j+1)/15, clipped
        int b = N_BINS - 1;
        #pragma unroll
        for (int j = N_BINS - 2; j >= 0; --j)
            if (conf <= (float)((j + 1) * (1.0 / N_BINS))) b = j;

        if (lane == 0) {
            atomicAdd(&hist[3 * b + 0], 1.0f);
            atomicAdd(&hist[3 * b + 1], conf);
            atomicAdd(&hist[3 * b + 2], acc);
        }
    }

    __syncthreads();
    for (int i = threadIdx.x; i < 3 * N_BINS; i += blockDim.x)
        if (hist[i] != 0.0f) atomicAdd(&ws[i], hist[i]);
}

__global__ void mce_final(const float* __restrict__ ws, float* __restrict__ out) {
    const int i = threadIdx.x;  // single wave of 32
    float gap = -INFINITY, prob = 0.0f, accu = 0.0f;
    if (i < N_BINS) {
        const float cnt = ws[3 * i + 0];
        if (cnt > 0.0f) {
            prob = ws[3 * i + 1] / cnt;
            accu = ws[3 * i + 2] / cnt;
            gap  = fabsf(prob - accu);
        }
    }
    float m = gap;
    #pragma unroll
    for (int off = 16; off; off >>= 1) m = fmaxf(m, __shfl_xor(m, off, 32));
    if (i == 0) out[0] = m;                 // mce
    if (i < N_BINS) {
        out[1 + i]          = prob;         // avg confidence per bin (0 if empty)
        out[1 + N_BINS + i] = accu;         // accuracy per bin (0 if empty)
    }
}

extern "C" void kernel_launch(void* const* d_in, const int* in_sizes, int n_in,
                              void* d_out, int out_size, void* d_ws, size_t ws_size,
                              hipStream_t stream) {
    const float* logits = (const float*)d_in[0];
    const int*   labels = (const int*)d_in[1];
    float* ws  = (float*)d_ws;
    float* out = (float*)d_out;
    const int n = in_sizes[1];  // number of rows (labels length)

    mce_init_ws<<<1, 64, 0, stream>>>(ws);
    mce_main<<<MAIN_BLOCKS, THREADS, 0, stream>>>(logits, labels, ws, n);
    mce_final<<<1, 32, 0, stream>>>(ws, out);
}